// NaiveSubMConv2d_9603546874113
// MI455X (gfx1250) — compile-verified
//
#include <hip/hip_runtime.h>

// Problem constants (match reference)
#define BB   4
#define HH   1024
#define WWID 1024
#define CIN  64
#define KOUT 64
#define NTAP 9
#define GRID_CELLS (BB * HH * WWID)

typedef __attribute__((ext_vector_type(2))) float v2f;
typedef __attribute__((ext_vector_type(8))) float v8f;

// Pointer types matching __builtin_amdgcn_global_load_async_to_lds_b128:
// arg0: global (AS1) pointer to 16-byte int vector; arg1: LDS (AS3) pointer.
typedef int v4i __attribute__((vector_size(16)));
typedef __attribute__((address_space(1))) v4i* gvec_ptr;
typedef __attribute__((address_space(3))) v4i* lvec_ptr;

#if __has_builtin(__builtin_amdgcn_global_load_async_to_lds_b128)
#define HAVE_ASYNC_COPY 1
#else
#define HAVE_ASYNC_COPY 0
#endif

static __device__ __forceinline__ void wait_async0() {
#if HAVE_ASYNC_COPY
#if __has_builtin(__builtin_amdgcn_s_wait_asynccnt)
    __builtin_amdgcn_s_wait_asynccnt(0);
#else
    asm volatile("s_wait_asynccnt 0x0" ::: "memory");
#endif
#endif
}

static __device__ __forceinline__ void async_copy16(const float* src, float* lds_dst) {
#if HAVE_ASYNC_COPY
    __builtin_amdgcn_global_load_async_to_lds_b128((gvec_ptr)src, (lvec_ptr)lds_dst,
                                                   0, 0);
#else
    const float4 v = *(const float4*)src;
    *(float4*)lds_dst = v;
#endif
}

// ---------------------------------------------------------------------------
// Kernel 1: fill the dense coordinate hash with -1, and zero the fallback row
// ---------------------------------------------------------------------------
__global__ void fill_grid_kernel(int* __restrict__ grid, float* __restrict__ zrow,
                                 int total) {
    int i = blockIdx.x * blockDim.x + threadIdx.x;
    if (i < total) grid[i] = -1;
    if (i < CIN) zrow[i] = 0.0f;
}

// ---------------------------------------------------------------------------
// Kernel 2: scatter row index at each active voxel
// ---------------------------------------------------------------------------
__global__ void scatter_grid_kernel(const int* __restrict__ idx,
                                    int* __restrict__ grid, int n) {
    int i = blockIdx.x * blockDim.x + threadIdx.x;
    if (i < n) {
        int b = idx[3 * i + 0];
        int h = idx[3 * i + 1];
        int w = idx[3 * i + 2];
        grid[b * (HH * WWID) + h * WWID + w] = i;
    }
}

// ---------------------------------------------------------------------------
// Kernel 3: one-time weight transpose  [K][3][3][C] -> wT[tap][c][k_out]
// so per-tap staging is a contiguous 16KB block (async-b128 friendly) and
// B-fragment LDS reads stay contiguous-across-lanes.
// ---------------------------------------------------------------------------
__global__ void transpose_weight_kernel(const float* __restrict__ w,
                                        float* __restrict__ wT) {
    int i = blockIdx.x * blockDim.x + threadIdx.x;  // linear over [K][9][C]
    if (i < KOUT * NTAP * CIN) {
        int ko  = i / (NTAP * CIN);
        int rem = i - ko * (NTAP * CIN);
        int tap = rem >> 6;
        int c   = rem & 63;
        wT[(tap * CIN + c) * KOUT + ko] = w[i];
    }
}

// ---------------------------------------------------------------------------
// Kernel 4: submanifold sparse conv via V_WMMA_F32_16X16X4_F32.
// 128 threads = 4 waves; each wave computes a 16-row x 64-col output tile.
// Weight slices are double-buffered in LDS and staged with async global->LDS
// copies (ASYNCcnt) that overlap the current tap's gathers + WMMAs.
//
// A-fragment (32-bit A 16x4, ISA 7.12.2): lane L: M=L&15, K-pair=(L>=16?2:0).
// B-fragment (4x16) mirrored. C/D: vgpr j holds row M=j+8*(L>=16), col N=L&15.
// ---------------------------------------------------------------------------
__launch_bounds__(128)
__global__ void subm_conv_wmma_kernel(const float* __restrict__ feat,
                                      const int*   __restrict__ idx,
                                      const float* __restrict__ wT,   // [9][C][K]
                                      const float* __restrict__ bias, // [K]
                                      const int*   __restrict__ grid,
                                      const float* __restrict__ zrow, // [C] zeros
                                      float*       __restrict__ out,  // [N][K]
                                      int n) {
    __shared__ float lw[2][CIN * KOUT];  // 2 x 16 KB double buffer

    const int lane = threadIdx.x & 31;
    const int wave = threadIdx.x >> 5;
    const int m    = lane & 15;   // row within tile / col within 16-block
    const int hi   = lane >> 4;   // 0: K-pair {0,1}, 1: K-pair {2,3}
    const int kb   = hi * 2;

    const int row_base = (blockIdx.x * 4 + wave) * 16;
    const int my_row   = row_base + m;
    const bool vrow    = my_row < n;
    const int crow     = vrow ? my_row : (n - 1);

    const int vb = idx[3 * crow + 0];
    const int vh = idx[3 * crow + 1];
    const int vw = idx[3 * crow + 2];

    v8f acc0 = {}, acc1 = {}, acc2 = {}, acc3 = {};

    // ---- stage tap 0 into buffer 0 ----
    for (int i = threadIdx.x; i < (CIN * KOUT) / 4; i += 128) {
        async_copy16(wT + i * 4, &lw[0][i * 4]);
    }
    wait_async0();
    __syncthreads();

    for (int tap = 0; tap < NTAP; ++tap) {
        const int cur = tap & 1;

        // ---- kick off async staging of the NEXT tap (overlaps compute) ----
        if (tap + 1 < NTAP) {
            const float* src = wT + (tap + 1) * (CIN * KOUT);
            float* dst = lw[cur ^ 1];
            for (int i = threadIdx.x; i < (CIN * KOUT) / 4; i += 128) {
                async_copy16(src + i * 4, dst + i * 4);
            }
        }

        // ---- neighbor lookup for this tap ----
        const int r  = tap / 3;
        const int s  = tap - r * 3;
        const int hh = vh + r - 1;
        const int ww = vw + s - 1;
        const bool inb = vrow && (hh >= 0) && (hh < HH) && (ww >= 0) && (ww < WWID);
        int nbr = -1;
        if (inb) nbr = grid[vb * (HH * WWID) + hh * WWID + ww];
        // Inactive lanes read the zero row: no per-K-step masking VALU,
        // EXEC stays all-1s as WMMA requires.
        const float* arow = (nbr >= 0) ? (feat + nbr * CIN) : zrow;
        const float* lwc  = lw[cur];

        // ---- 16 K-steps of depth 4 over the 64 input channels ----
        for (int kk = 0; kk < 16; ++kk) {
            const int cb = kk * 4 + kb;

            const float2 a2 = *(const float2*)(arow + cb);
            v2f A;
            A.x = a2.x;
            A.y = a2.y;

            v2f B0, B1, B2, B3;
            B0.x = lwc[cb * KOUT + m];
            B0.y = lwc[(cb + 1) * KOUT + m];
            B1.x = lwc[cb * KOUT + 16 + m];
            B1.y = lwc[(cb + 1) * KOUT + 16 + m];
            B2.x = lwc[cb * KOUT + 32 + m];
            B2.y = lwc[(cb + 1) * KOUT + 32 + m];
            B3.x = lwc[cb * KOUT + 48 + m];
            B3.y = lwc[(cb + 1) * KOUT + 48 + m];

            acc0 = __builtin_amdgcn_wmma_f32_16x16x4_f32(
                false, A, false, B0, (short)0, acc0, false, false);
            acc1 = __builtin_amdgcn_wmma_f32_16x16x4_f32(
                false, A, false, B1, (short)0, acc1, false, false);
            acc2 = __builtin_amdgcn_wmma_f32_16x16x4_f32(
                false, A, false, B2, (short)0, acc2, false, false);
            acc3 = __builtin_amdgcn_wmma_f32_16x16x4_f32(
                false, A, false, B3, (short)0, acc3, false, false);
        }

        // ---- next tap's weights must be resident before anyone reads them ----
        wait_async0();
        __syncthreads();
    }

    // ---- epilogue: C/D layout -> global, add bias, guard N tail ----
    const float bv0 = bias[m];
    const float bv1 = bias[16 + m];
    const float bv2 = bias[32 + m];
    const float bv3 = bias[48 + m];
    for (int j = 0; j < 8; ++j) {
        const int gr = row_base + j + hi * 8;
        if (gr < n) {
            float* o = out + gr * KOUT;
            o[m]      = acc0[j] + bv0;
            o[16 + m] = acc1[j] + bv1;
            o[32 + m] = acc2[j] + bv2;
            o[48 + m] = acc3[j] + bv3;
        }
    }
}

// ---------------------------------------------------------------------------
extern "C" void kernel_launch(void* const* d_in, const int* in_sizes, int n_in,
                              void* d_out, int out_size, void* d_ws, size_t ws_size,
                              hipStream_t stream) {
    const float* feat   = (const float*)d_in[0];
    const int*   idx    = (const int*)d_in[1];
    const float* weight = (const float*)d_in[2];
    const float* bias   = (const float*)d_in[3];
    float*       out    = (float*)d_out;

    const int n = in_sizes[1] / 3;  // indices is [N,3]

    // Workspace layout: [0,16MB) dense hash | +16MB zero row | +16MB+4KB wT
    char* ws = (char*)d_ws;
    int*   grid = (int*)ws;
    float* zrow = (float*)(ws + (size_t)GRID_CELLS * sizeof(int));
    float* wT   = (float*)(ws + (size_t)GRID_CELLS * sizeof(int) + 4096);

    fill_grid_kernel<<<(GRID_CELLS + 255) / 256, 256, 0, stream>>>(grid, zrow,
                                                                   GRID_CELLS);
    scatter_grid_kernel<<<(n + 255) / 256, 256, 0, stream>>>(idx, grid, n);
    transpose_weight_kernel<<<(KOUT * NTAP * CIN + 255) / 256, 256, 0, stream>>>(
        weight, wT);

    const int tiles  = (n + 15) / 16;
    const int blocks = (tiles + 3) / 4;  // 4 waves (tiles) per 128-thread block
    subm_conv_wmma_kernel<<<blocks, 128, 0, stream>>>(feat, idx, wT, bias, grid,
                                                      zrow, out, n);
}